// QueryAndGroupPN_38044820308017
// MI455X (gfx1250) — compile-verified
//
#include <hip/hip_runtime.h>

// QueryAndGroupPN for MI455X (gfx1250, wave32).
// B=8, N=4096, Np=1024, C=256, nsample=32, radius=0.2
//
// Kernel 1: ball query via V_WMMA_F32_16X16X4_F32 (d2 = |q|^2+|p|^2 - 2 q.p as a
//           rank-3 GEMM tile), wave32 ballot-based ordered index selection.
// Kernel 2: per-query mode-pose (shfl/ballot reductions) + rotated xyz channels.
// Kernel 3: store-bandwidth-bound feature gather (coalesced 32-lane stores,
//           gathers are L2-resident: 4MB feature matrix per batch vs 192MB L2).

#define B_   8
#define N_   4096
#define NP_  1024
#define C_   256
#define NS_  32
#define RADIUS_ 0.2f

typedef __attribute__((ext_vector_type(2))) float v2f;
typedef __attribute__((ext_vector_type(8))) float v8f;

__device__ __forceinline__ unsigned ballot32(bool p) {
    return __builtin_amdgcn_ballot_w32(p);
}

// ---------------------------------------------------------------------------
// Kernel 1: ball query. One wave handles 16 query points; scans N in 16-point
// tiles. A (16x4, f32) = -2*q (z in K=2 slot, K=3 zero), B (4x16) = p coords,
// C = |q|^2 + |p|^2  =>  D = d2 tile, striped per the CDNA5 C/D layout
// (VGPR r: lanes 0-15 -> row r, lanes 16-31 -> row r+8).
// ---------------------------------------------------------------------------
__global__ void __launch_bounds__(32)
ball_query_wmma(const float* __restrict__ xyz, const float* __restrict__ new_xyz,
                int* __restrict__ idx_out) {
    const int lane = threadIdx.x;
    const int g  = blockIdx.x;                 // B * NP/16 groups
    const int b  = g / (NP_ / 16);
    const int q0 = (g % (NP_ / 16)) * 16;
    const int m  = lane & 15;
    const float r2 = RADIUS_ * RADIUS_;

    // Query point for row m (both lane halves load the same row m).
    const float* qp = new_xyz + ((size_t)b * NP_ + (q0 + m)) * 3;
    const float qx = qp[0], qy = qp[1], qz = qp[2];
    const float q2m = qx * qx + qy * qy + qz * qz;

    // A operand: lanes 0-15 carry K=0,1 ; lanes 16-31 carry K=2,3. Fold -2 in.
    v2f A;
    if (lane < 16) { A.x = -2.0f * qx; A.y = -2.0f * qy; }
    else           { A.x = -2.0f * qz; A.y = 0.0f;       }

    // |q|^2 broadcast per accumulator VGPR row (row r low half, row r+8 high).
    float q2row[8];
#pragma unroll
    for (int r = 0; r < 8; ++r)
        q2row[r] = __shfl(q2m, (lane < 16) ? r : (r + 8), 32);

    int cnt = 0, first = 0;
    int* myIdx = idx_out + ((size_t)b * NP_ + (q0 + m)) * NS_;
    const float* xb = xyz + (size_t)b * N_ * 3;

    for (int nbase = 0; nbase < N_; nbase += 16) {
        const int n = nbase + m;
        const float* pp = xb + (size_t)n * 3;
        const float px = pp[0], py = pp[1], pz = pp[2];
        const float p2 = px * px + py * py + pz * pz;

        v2f Bv;
        if (lane < 16) { Bv.x = px; Bv.y = py; }
        else           { Bv.x = pz; Bv.y = 0.0f; }

        v8f Cv;
#pragma unroll
        for (int r = 0; r < 8; ++r) Cv[r] = q2row[r] + p2;

        // d2[16x16] tile on the matrix core (EXEC is all-ones here).
        v8f d2 = __builtin_amdgcn_wmma_f32_16x16x4_f32(
            false, A, false, Bv, (short)0, Cv, false, false);

        // Per-row 16-bit "within radius" masks. Lane r (r<8) keeps the ballot
        // for accumulator VGPR r (rows r and r+8).
        unsigned keep = 0u;
#pragma unroll
        for (int r = 0; r < 8; ++r) {
            unsigned mm = ballot32(d2[r] < r2);
            if (lane == r) keep = mm;
        }
        unsigned full    = (unsigned)__shfl((int)keep, m & 7, 32);
        unsigned rowMask = (m < 8) ? (full & 0xFFFFu) : (full >> 16);

        // Lanes 0-15 own one query row each: append indices in ascending order.
        if (lane < 16) {
            unsigned msk = rowMask;
            while (msk && cnt < NS_) {
                int bit = __ffs(msk) - 1;
                msk &= msk - 1;
                int id = nbase + bit;
                if (cnt == 0) first = id;
                myIdx[cnt++] = id;
            }
        }
        // Uniform early exit once every row has nsample hits.
        unsigned done = ballot32((lane < 16) ? (cnt >= NS_) : true);
        if (done == 0xFFFFFFFFu) break;
    }

    if (lane < 16) {
        if (cnt == 0) first = 0;           // no neighbor -> index 0 (reference)
        for (int j = cnt; j < NS_; ++j) myIdx[j] = first;
    }
}

// ---------------------------------------------------------------------------
// Kernel 2: one wave per query point: mode-pose via shfl/ballot reductions,
// then rotated grouped-xyz channels 0..2.
// ---------------------------------------------------------------------------
__global__ void __launch_bounds__(256)
mode_pose_rot(const float* __restrict__ xyz, const float* __restrict__ new_xyz,
              const float* __restrict__ pose, const unsigned char* __restrict__ pmask,
              const int* __restrict__ idx_in, float* __restrict__ outF) {
    const int lane = threadIdx.x & 31;
    const int wid  = threadIdx.x >> 5;
    const int q    = blockIdx.x * 8 + wid;      // 0 .. B*NP-1
    const int b    = q / NP_;
    const int np   = q % NP_;
    const int k    = lane;

    const int id  = idx_in[(size_t)q * NS_ + k];
    const int id0 = __shfl(id, 0, 32);
    const bool filler = (k > 0) && (id == id0);

    const float pv = pose[(size_t)b * N_ + id];
    const bool  mk = pmask[(size_t)b * N_ + id] != 0;

    // Hash-based unique fillers (stand-in for jax PRNG; uniqueness is what
    // matters for the mode computation).
    unsigned h = (unsigned)(q * NS_ + k) * 2654435761u;
    h ^= h >> 13; h *= 0x85ebca6bu; h ^= h >> 16;
    const float r1 = (float)(h >> 8) * (1.0f / 16777216.0f);
    unsigned h2 = h * 0x27d4eb2du; h2 ^= h2 >> 15;
    const float r2v = (float)(h2 >> 8) * (1.0f / 16777216.0f);

    float cl = mk ? pv : (7.0f + r1);
    if (filler) cl = 11.0f + r2v;
    const bool clm = mk && !filler;

    // counts[k] = #{j : cl[j] == cl[k]}
    int count = 0;
#pragma unroll
    for (int j = 0; j < NS_; ++j)
        count += (__shfl(cl, j, 32) == cl) ? 1 : 0;

    int maxc = count;
#pragma unroll
    for (int o = 16; o > 0; o >>= 1) maxc = max(maxc, __shfl_xor(maxc, o, 32));

    float mode = (count == maxc) ? cl : __builtin_inff();
#pragma unroll
    for (int o = 16; o > 0; o >>= 1) mode = fminf(mode, __shfl_xor(mode, o, 32));

    const unsigned eqm = ballot32(cl == mode);
    const int posn = __ffs(eqm) - 1;
    const int fg = __shfl((int)clm, posn, 32);
    mode = fg ? mode : 0.0f;

    if (lane == 0)
        outF[(size_t)B_ * 259 * NP_ * NS_ + (size_t)b * NP_ + np] = mode;

    // Rotated grouped xyz -> channels 0..2 of new_features.
    const float* pp = xyz + ((size_t)b * N_ + id) * 3;
    const float* qp = new_xyz + ((size_t)b * NP_ + np) * 3;
    const float gx = (pp[0] - qp[0]) * (1.0f / RADIUS_);
    const float gy = (pp[1] - qp[1]) * (1.0f / RADIUS_);
    const float gz = (pp[2] - qp[2]) * (1.0f / RADIUS_);

    float sA, cA;
    __sincosf(-mode, &sA, &cA);

    const size_t chs  = (size_t)NP_ * NS_;
    const size_t base = (((size_t)b * 259 + 0) * NP_ + np) * NS_ + k;
    outF[base]           = cA * gx - sA * gy;
    outF[base + chs]     = sA * gx + cA * gy;
    outF[base + 2 * chs] = gz;
}

// ---------------------------------------------------------------------------
// Kernel 3: feature gather, channels 3..258. 8 waves/block, wave w streams
// channels w, w+8, ... 32 contiguous lanes per channel -> 128B coalesced
// stores; gathers hit the (L2-resident) 4MB per-batch feature matrix.
// ---------------------------------------------------------------------------
__global__ void __launch_bounds__(256)
gather_features(const float* __restrict__ feat, const int* __restrict__ idx_in,
                float* __restrict__ outF) {
    const int t  = threadIdx.x;
    const int k  = t & 31;
    const int c0 = t >> 5;                      // 0..7
    const int q  = blockIdx.x;                  // 0 .. B*NP-1
    const int b  = q / NP_;
    const int np = q % NP_;

    const int id = idx_in[(size_t)q * NS_ + k];
    const float* fb = feat + (size_t)b * C_ * N_;
    const size_t chs   = (size_t)NP_ * NS_;
    const size_t obase = (((size_t)b * 259 + 3) * NP_ + np) * NS_ + k;

#pragma unroll 4
    for (int c = c0; c < C_; c += 8)
        outF[obase + (size_t)c * chs] = fb[(size_t)c * N_ + id];
}

// ---------------------------------------------------------------------------
extern "C" void kernel_launch(void* const* d_in, const int* in_sizes, int n_in,
                              void* d_out, int out_size, void* d_ws, size_t ws_size,
                              hipStream_t stream) {
    const float*         xyz     = (const float*)d_in[0];          // (8,4096,3)
    const float*         new_xyz = (const float*)d_in[1];          // (8,1024,3)
    const float*         feat    = (const float*)d_in[2];          // (8,256,4096)
    const float*         pose    = (const float*)d_in[3];          // (8,4096)
    const unsigned char* pmask   = (const unsigned char*)d_in[4];  // (8,4096) bool
    float* outF   = (float*)d_out;
    int*   idx_ws = (int*)d_ws;                                    // 8*1024*32 ints = 1MB

    ball_query_wmma<<<dim3(B_ * NP_ / 16), dim3(32), 0, stream>>>(xyz, new_xyz, idx_ws);
    mode_pose_rot<<<dim3(B_ * NP_ / 8), dim3(256), 0, stream>>>(xyz, new_xyz, pose,
                                                                pmask, idx_ws, outF);
    gather_features<<<dim3(B_ * NP_), dim3(256), 0, stream>>>(feat, idx_ws, outF);
}